// BatchAllTripletLoss_25486335934653
// MI455X (gfx1250) — compile-verified
//
#include <hip/hip_runtime.h>
#include <hip/hip_bf16.h>

typedef __attribute__((ext_vector_type(2))) float v2f;
typedef __attribute__((ext_vector_type(8))) float v8f;

#define BSZ 512
#define DIM 128
#define MARGIN_F 0.2f
#define EPS_F 1e-16f

// ---------------------------------------------------------------------------
// Kernel 1: squared L2 norm of each embedding row. 512 rows, 1 thread/row.
// ---------------------------------------------------------------------------
__global__ void tl_sqnorm_kernel(const float* __restrict__ E, float* __restrict__ sq) {
    int r = blockIdx.x * blockDim.x + threadIdx.x;
    if (r < BSZ) {
        float s = 0.0f;
        #pragma unroll 4
        for (int k = 0; k < DIM; ++k) {
            float v = E[r * DIM + k];
            s = fmaf(v, v, s);
        }
        sq[r] = s;
    }
}

// ---------------------------------------------------------------------------
// Kernel 2: Gram matrix via V_WMMA_F32_16X16X4_F32 (full fp32 matrix pipe),
// fused into the distance transform d[i,j] = sqrt(relu(sq_i - 2g + sq_j)).
// One wave (32 lanes) per 16x16 tile of the 512x512 output: 32x32 = 1024 tiles.
// K loop: 128/4 = 32 WMMA ops per tile.
//
// A (16x4 f32) lane layout (ISA 7.12.2): lanes 0-15 hold K=0,1 of row M=lane;
// lanes 16-31 hold K=2,3 of row M=lane-16. B (4x16) mirrors this with columns.
// C/D: VGPR r holds row r (lanes 0-15) and row r+8 (lanes 16-31), N = lane%16.
// ---------------------------------------------------------------------------
__global__ void tl_gram_dist_kernel(const float* __restrict__ E,
                                    const float* __restrict__ sq,
                                    float* __restrict__ dmat) {
    const int tile = blockIdx.x;          // 0..1023
    const int ti = tile >> 5;             // tile row  (i block)
    const int tj = tile & 31;             // tile col  (j block)
    const int i0 = ti * 16;
    const int j0 = tj * 16;

    const int lane = threadIdx.x;         // 0..31, full wave (EXEC all ones)
    const int half = lane >> 4;           // 0 -> K pair (0,1), 1 -> K pair (2,3)
    const int m    = lane & 15;           // row within block for A/B fragments

    const float* rowA = E + (i0 + m) * DIM;   // A-fragment source row
    const float* rowB = E + (j0 + m) * DIM;   // B-fragment source row

    v8f c = {};
    for (int k0 = 0; k0 < DIM; k0 += 4) {
        const int k = k0 + half * 2;
        v2f a, b;
        a.x = rowA[k];
        a.y = rowA[k + 1];
        b.x = rowB[k];
        b.y = rowB[k + 1];
        // (neg_a, A, neg_b, B, c_mod, C, reuse_a, reuse_b)
        c = __builtin_amdgcn_wmma_f32_16x16x4_f32(
                false, a, false, b, (short)0, c, false, false);
    }

    const int n   = lane & 15;
    const float sqj = sq[j0 + n];
    #pragma unroll
    for (int r = 0; r < 8; ++r) {
        const int i = i0 + r + 8 * half;
        const float g   = c[r];
        float dsq = sq[i] - 2.0f * g + sqj;
        dsq = fmaxf(dsq, 0.0f);
        dmat[i * BSZ + (j0 + n)] = (dsq > 0.0f) ? sqrtf(dsq) : 0.0f;
    }
}

// ---------------------------------------------------------------------------
// Kernel 3: per-anchor triplet accumulation. One block per anchor a.
// valid(a,p,n) <=> p!=a, id_p==id_a, id_n!=id_a  (p!=n, n!=a implied).
// sum += relu(d[a,p] - d[a,n] + margin); cnt += (value > EPS).
// Positive list built serially by thread 0 -> deterministic float order.
// ---------------------------------------------------------------------------
__global__ void tl_triplet_kernel(const float* __restrict__ dmat,
                                  const int* __restrict__ ids,
                                  float* __restrict__ blocksum,
                                  unsigned int* __restrict__ blockcnt) {
    __shared__ float drow[BSZ];
    __shared__ int   idsh[BSZ];
    __shared__ int   poslist[BSZ];
    __shared__ int   pcount;
    __shared__ float        redf[256];
    __shared__ unsigned int redu[256];

    const int a = blockIdx.x;
    const int t = threadIdx.x;
    const int ida = ids[a];

    for (int j = t; j < BSZ; j += 256) {
        drow[j] = dmat[a * BSZ + j];
        idsh[j] = ids[j];
    }
    __syncthreads();

    if (t == 0) {
        int c = 0;
        for (int j = 0; j < BSZ; ++j)
            if (j != a && idsh[j] == ida) poslist[c++] = j;
        pcount = c;
    }
    __syncthreads();

    float sum = 0.0f;
    unsigned int cnt = 0;
    const int np = pcount;
    for (int pi = 0; pi < np; ++pi) {
        const float dpm = drow[poslist[pi]] + MARGIN_F;
        for (int n = t; n < BSZ; n += 256) {
            if (idsh[n] != ida) {
                const float v = dpm - drow[n];
                if (v > 0.0f) {
                    sum += v;
                    if (v > EPS_F) ++cnt;
                }
            }
        }
    }

    redf[t] = sum;
    redu[t] = cnt;
    __syncthreads();
    for (int s = 128; s > 0; s >>= 1) {
        if (t < s) { redf[t] += redf[t + s]; redu[t] += redu[t + s]; }
        __syncthreads();
    }
    if (t == 0) { blocksum[a] = redf[0]; blockcnt[a] = redu[0]; }
}

// ---------------------------------------------------------------------------
// Kernel 4: deterministic final reduction over 512 anchor partials.
// ---------------------------------------------------------------------------
__global__ void tl_finalize_kernel(const float* __restrict__ blocksum,
                                   const unsigned int* __restrict__ blockcnt,
                                   float* __restrict__ out) {
    __shared__ float        rf[BSZ];
    __shared__ unsigned int ru[BSZ];
    const int t = threadIdx.x;
    rf[t] = blocksum[t];
    ru[t] = blockcnt[t];
    __syncthreads();
    for (int s = 256; s > 0; s >>= 1) {
        if (t < s) { rf[t] += rf[t + s]; ru[t] += ru[t + s]; }
        __syncthreads();
    }
    if (t == 0) out[0] = rf[0] / ((float)ru[0] + EPS_F);
}

// ---------------------------------------------------------------------------
extern "C" void kernel_launch(void* const* d_in, const int* in_sizes, int n_in,
                              void* d_out, int out_size, void* d_ws, size_t ws_size,
                              hipStream_t stream) {
    const float* embs = (const float*)d_in[0];
    const int*   ids  = (const int*)d_in[1];
    float* out = (float*)d_out;

    char* ws = (char*)d_ws;
    float*        dmat     = (float*)(ws);                              // 512*512*4 = 1 MB
    float*        sq       = (float*)(ws + (size_t)BSZ * BSZ * 4);      // 2 KB
    float*        blocksum = (float*)(ws + (size_t)BSZ * BSZ * 4 + 2048);
    unsigned int* blockcnt = (unsigned int*)(ws + (size_t)BSZ * BSZ * 4 + 4096);

    tl_sqnorm_kernel<<<2, 256, 0, stream>>>(embs, sq);
    tl_gram_dist_kernel<<<(BSZ / 16) * (BSZ / 16), 32, 0, stream>>>(embs, sq, dmat);
    tl_triplet_kernel<<<BSZ, 256, 0, stream>>>(dmat, ids, blocksum, blockcnt);
    tl_finalize_kernel<<<1, BSZ, 0, stream>>>(blocksum, blockcnt, out);
}